// AttentionAggregator_88545045774530
// MI455X (gfx1250) — compile-verified
//
#include <hip/hip_runtime.h>

typedef __attribute__((ext_vector_type(2))) float v2f;
typedef __attribute__((ext_vector_type(8))) float v8f;

#define DIM        128
#define KNEIGH     32
#define OUTD       128
#define NODES_PB   16        // nodes per block (M of the WMMA GEMM)
#define CTX_STRIDE 132       // 128 + 4 pad: stride%64==4 -> conflict-free A reads

// One online-softmax step: dot(self,row) -> wave-reduce -> update (m, l, acc)
__device__ __forceinline__ void online_step(const float4& sv, const float4& rv,
                                            float& m, float& l, float4& acc) {
  float part = sv.x * rv.x + sv.y * rv.y + sv.z * rv.z + sv.w * rv.w;
  part += __shfl_xor(part, 16, 32);
  part += __shfl_xor(part, 8, 32);
  part += __shfl_xor(part, 4, 32);
  part += __shfl_xor(part, 2, 32);
  part += __shfl_xor(part, 1, 32);           // all 32 lanes now hold the score
  float mn    = fmaxf(m, part);
  float scale = __expf(m - mn);              // v_exp_f32
  float p     = __expf(part - mn);
  l = l * scale + p;
  acc.x = fmaf(acc.x, scale, p * rv.x);
  acc.y = fmaf(acc.y, scale, p * rv.y);
  acc.z = fmaf(acc.z, scale, p * rv.z);
  acc.w = fmaf(acc.w, scale, p * rv.w);
  m = mn;
}

__global__ __launch_bounds__(256) void attn_agg_kernel(
    const float* __restrict__ self_vecs,    // [N,128]
    const float* __restrict__ neigh_vecs,   // [N,32,128]
    const float* __restrict__ W,            // [128,128]
    float* __restrict__ out,                // [N,128]
    int N)
{
  __shared__ float ctx[NODES_PB * CTX_STRIDE];   // 8.25 KB context staging

  const int tid       = threadIdx.x;
  const int lane      = tid & 31;
  const int wave      = tid >> 5;                // 0..7
  const int blockBase = blockIdx.x * NODES_PB;

  // ---------------- Phase 1: fused scores + online softmax + context ------
  // Each wave processes 2 nodes; single pass over the neighbor stream.
  #pragma unroll
  for (int iter = 0; iter < 2; ++iter) {
    const int nloc = wave + iter * 8;
    const int node = blockBase + nloc;
    float* crow = ctx + nloc * CTX_STRIDE + lane * 4;
    if (node < N) {
      const float4* sptr = (const float4*)(self_vecs + (size_t)node * DIM);
      const float4  sv   = sptr[lane];           // this lane's 4-dim slice

      float  m   = -3.402823466e+38f;
      float  l   = 0.0f;
      float4 acc = make_float4(0.f, 0.f, 0.f, 0.f);

      const float4* nptr =
          (const float4*)(neigh_vecs + (size_t)node * KNEIGH * DIM);
      #pragma unroll 4
      for (int k = 0; k < KNEIGH; ++k) {
        float4 rv = nptr[k * 32 + lane];         // coalesced 512B per row
        online_step(sv, rv, m, l, acc);
      }
      // appended self row (reference concatenates query last)
      online_step(sv, sv, m, l, acc);

      const float inv = 1.0f / l;
      *(float4*)crow = make_float4(acc.x * inv, acc.y * inv,
                                   acc.z * inv, acc.w * inv);
    } else {
      *(float4*)crow = make_float4(0.f, 0.f, 0.f, 0.f);
    }
  }
  __syncthreads();

  // ---------------- Phase 2: (16 x 128) @ (128 x 128) via f32 WMMA --------
  // Wave w owns output columns [16w, 16w+16). K=128 -> 32 x wmma 16x16x4.
  const int n0   = wave * 16;
  const int half = lane >> 4;     // 0: lanes 0-15, 1: lanes 16-31
  const int l16  = lane & 15;

  v8f c = {};
  #pragma unroll 4
  for (int kk = 0; kk < 32; ++kk) {
    const int k0 = kk * 4 + half * 2;    // A/B lane-half K offset per ISA layout
    // A fragment (16x4 f32): lanes 0-15 hold K={k0,k0+1}, M=lane;
    //                        lanes 16-31 hold K={k0+2,k0+3}, M=lane-16
    const float* ap = ctx + l16 * CTX_STRIDE + k0;
    v2f a;
    a.x = ap[0];
    a.y = ap[1];
    // B fragment (4x16 f32): VGPR0 lanes0-15 K=k0 / lanes16-31 K=k0+2; VGPR1 +1
    const float* bp = W + (size_t)k0 * OUTD + n0 + l16;
    v2f b;
    b.x = bp[0];
    b.y = bp[OUTD];
    c = __builtin_amdgcn_wmma_f32_16x16x4_f32(
        /*neg_a=*/false, a, /*neg_b=*/false, b,
        /*c_mod=*/(short)0, c, /*reuse_a=*/false, /*reuse_b=*/false);
  }

  // C/D layout: VGPR v -> M = v (lanes 0-15) or v+8 (lanes 16-31); N = l16
  #pragma unroll
  for (int v = 0; v < 8; ++v) {
    const int mrow = v + half * 8;
    const int node = blockBase + mrow;
    if (node < N) {
      out[(size_t)node * OUTD + n0 + l16] = fmaxf(c[v], 0.0f);
    }
  }
}

extern "C" void kernel_launch(void* const* d_in, const int* in_sizes, int n_in,
                              void* d_out, int out_size, void* d_ws, size_t ws_size,
                              hipStream_t stream) {
  const float* self_vecs  = (const float*)d_in[0];
  const float* neigh_vecs = (const float*)d_in[1];
  const float* W          = (const float*)d_in[2];
  float* out = (float*)d_out;

  const int N    = in_sizes[0] / DIM;                  // 100000
  const int grid = (N + NODES_PB - 1) / NODES_PB;      // 6250

  attn_agg_kernel<<<grid, 256, 0, stream>>>(self_vecs, neigh_vecs, W, out, N);
}